// MaxFusion_35862976922368
// MI455X (gfx1250) — compile-verified
//
#include <hip/hip_runtime.h>
#include <hip/hip_bf16.h>
#include <math.h>

// Problem constants (match reference)
constexpr int B_ = 4, L_ = 5, C_ = 64, H_ = 128, W_ = 256;
constexpr int HW_   = H_ * W_;          // 32768 elements
constexpr int CHW_  = C_ * HW_;         // 2,097,152 elements (8 MiB)
constexpr int NTOT_ = 16;               // x.shape[0]
constexpr int CPB_  = 16;               // channels per block (IOFFSET-folded unroll)
constexpr float VOXEL_ = 0.4f;
constexpr float DS_    = 2.0f;

// One block = one output row (256 lanes = 256 w), 16 channels, one b.
// grid = (H, C/CPB, B), block = (W) = 256 threads = 8 wave32s.
__global__ __launch_bounds__(W_) void warp_max_fuse_kernel(
    const float* __restrict__ x,          // (16, C, H, W)
    const int*   __restrict__ record_len, // (B,)
    const float* __restrict__ tmat,       // (B, L, L, 4, 4)
    float*       __restrict__ out)        // (B, C, H, W)
{
    const int w  = threadIdx.x;
    const int h  = blockIdx.x;
    const int c0 = blockIdx.y * CPB_;
    const int b  = blockIdx.z;

    // ---- block-uniform record bookkeeping (force into SGPRs) ----
    int off = 0;
    #pragma unroll
    for (int bb = 0; bb < B_; ++bb) off += (bb < b) ? record_len[bb] : 0;
    int nrec = record_len[b];
    nrec = nrec < L_ ? nrec : L_;
    nrec = __builtin_amdgcn_readfirstlane(nrec);
    off  = __builtin_amdgcn_readfirstlane(off);

    // normalized output-grid coordinate (jnp.linspace(-1,1,N))
    const float gx = -1.0f + (float)w * (2.0f / (float)(W_ - 1));
    const float gy = -1.0f + (float)h * (2.0f / (float)(H_ - 1));

    // ---- per-agent bilinear sampling state (computed once, reused 16x) ----
    const float* basep[L_];   // scalar base pointers (SGPR pairs after readfirstlane)
    int   voff[L_][4];        // element offsets of the 4 corners within (H,W)
    float wgt [L_][4];        // corner weights with OOB validity folded to 0

    #pragma unroll
    for (int l = 0; l < L_; ++l) {
        int n = off + l;
        n = n < (NTOT_ - 1) ? n : (NTOT_ - 1);           // jnp.clip upper bound
        n = __builtin_amdgcn_readfirstlane(n);
        basep[l] = x + (size_t)n * CHW_ + (size_t)c0 * HW_;

        // theta = normalize(t[b][0][l]) : rows {0,1}, cols {0,1,3}
        const float* T = tmat + ((size_t)b * L_ * L_ + (size_t)l) * 16;
        const float m00 = T[0];
        const float m01 = T[1] * ((float)H_ / (float)W_);
        const float m02 = T[3] * (2.0f / (DS_ * VOXEL_ * (float)W_));
        const float m10 = T[4] * ((float)W_ / (float)H_);
        const float m11 = T[5];
        const float m12 = T[7] * (2.0f / (DS_ * VOXEL_ * (float)H_));

        const float grid_x = m00 * gx + m01 * gy + m02;
        const float grid_y = m10 * gx + m11 * gy + m12;

        const float ix = (grid_x + 1.0f) * 0.5f * (float)(W_ - 1);
        const float iy = (grid_y + 1.0f) * 0.5f * (float)(H_ - 1);

        const float x0f = floorf(ix), y0f = floorf(iy);
        const float x1f = x0f + 1.0f, y1f = y0f + 1.0f;
        const float wx1 = ix - x0f,   wx0 = 1.0f - wx1;
        const float wy1 = iy - y0f,   wy0 = 1.0f - wy1;

        const float xs [2] = {x0f, x1f};
        const float ys [2] = {y0f, y1f};
        const float wxs[2] = {wx0, wx1};
        const float wys[2] = {wy0, wy1};

        #pragma unroll
        for (int cy = 0; cy < 2; ++cy) {
            #pragma unroll
            for (int cx = 0; cx < 2; ++cx) {
                const float xc = xs[cx], yc = ys[cy];
                const bool valid = (xc >= 0.0f) && (xc <= (float)(W_ - 1)) &&
                                   (yc >= 0.0f) && (yc <= (float)(H_ - 1));
                float xcl = xc < 0.0f ? 0.0f : (xc > (float)(W_ - 1) ? (float)(W_ - 1) : xc);
                float ycl = yc < 0.0f ? 0.0f : (yc > (float)(H_ - 1) ? (float)(H_ - 1) : yc);
                const int xi = (int)xcl;
                const int yi = (int)ycl;
                voff[l][cy * 2 + cx] = yi * W_ + xi;
                wgt [l][cy * 2 + cx] = valid ? (wxs[cx] * wys[cy]) : 0.0f;
            }
        }
        // gfx1250 streaming prefetch of this agent's gather region
        __builtin_prefetch(basep[l] + voff[l][0], 0, 1);
    }

    // ---- gather + bilinear + running max over active agents ----
    float acc[CPB_];
    #pragma unroll
    for (int k = 0; k < CPB_; ++k) acc[k] = -__builtin_huge_valf();

    #pragma unroll
    for (int l = 0; l < L_; ++l) {
        if (l < nrec) {                      // scalar branch: skip inactive agents
            const float* bp  = basep[l];
            const int   o00 = voff[l][0], o01 = voff[l][1],
                        o10 = voff[l][2], o11 = voff[l][3];
            const float w00 = wgt[l][0], w01 = wgt[l][1],
                        w10 = wgt[l][2], w11 = wgt[l][3];
            #pragma unroll
            for (int k = 0; k < CPB_; ++k) {
                const int co = k * HW_;      // constant -> folds into IOFFSET
                float v =      bp[o00 + co] * w00;
                v = fmaf(bp[o01 + co], w01, v);
                v = fmaf(bp[o10 + co], w10, v);
                v = fmaf(bp[o11 + co], w11, v);
                acc[k] = fmaxf(acc[k], v);
            }
        }
    }

    float* op = out + ((size_t)b * C_ + (size_t)c0) * HW_ + (size_t)h * W_ + w;
    #pragma unroll
    for (int k = 0; k < CPB_; ++k) op[k * HW_] = acc[k];
}

extern "C" void kernel_launch(void* const* d_in, const int* in_sizes, int n_in,
                              void* d_out, int out_size, void* d_ws, size_t ws_size,
                              hipStream_t stream) {
    (void)in_sizes; (void)n_in; (void)out_size; (void)d_ws; (void)ws_size;
    const float* x    = (const float*)d_in[0];
    const int*   rlen = (const int*)d_in[1];
    const float* tmat = (const float*)d_in[2];
    float*       out  = (float*)d_out;

    dim3 grid(H_, C_ / CPB_, B_);   // (128, 4, 4) = 2048 blocks
    dim3 block(W_);                  // 256 threads = 8 wave32s
    warp_max_fuse_kernel<<<grid, block, 0, stream>>>(x, rlen, tmat, out);
}